// TransREmbedding_76605036691532
// MI455X (gfx1250) — compile-verified
//
#include <hip/hip_runtime.h>
#include <hip/hip_bf16.h>
#include <math.h>

typedef __attribute__((ext_vector_type(2))) float v2f;
typedef __attribute__((ext_vector_type(8))) float v8f;

#define NUM_REL   1000
#define BATCH     8192
#define DIM       256
#define MAX_TILES (BATCH / 16 + NUM_REL)   // sum ceil(c_r/16) <= 512 + 1000 = 1512
#define COL_STRIDE 260                     // 256 + 4 pad: conflict-free LDS B-fragment reads

// ---------------- prep kernels (relation grouping in d_ws) ----------------

__global__ void k_zero(int* counts, int* ntiles) {
    int i = blockIdx.x * blockDim.x + threadIdx.x;
    if (i < NUM_REL) counts[i] = 0;
    if (i == 0) *ntiles = 0;
}

__global__ void k_hist(const int* __restrict__ r, int* counts) {
    int i = blockIdx.x * blockDim.x + threadIdx.x;
    if (i < BATCH) atomicAdd(&counts[r[i]], 1);
}

// single-thread scan over 1000 relations: offsets + tile descriptors
__global__ void k_build(const int* __restrict__ counts, int* cursors,
                        int* tiles, int* ntiles) {
    int acc = 0, nt = 0;
    for (int rr = 0; rr < NUM_REL; ++rr) {
        int c = counts[rr];
        cursors[rr] = acc;
        for (int s = 0; s < c; s += 16) {
            int len = c - s; if (len > 16) len = 16;
            tiles[nt * 3 + 0] = rr;
            tiles[nt * 3 + 1] = acc + s;
            tiles[nt * 3 + 2] = len;
            ++nt;
        }
        acc += c;
    }
    *ntiles = nt;
}

__global__ void k_scatter(const int* __restrict__ r, int* cursors, int* perm) {
    int i = blockIdx.x * blockDim.x + threadIdx.x;
    if (i < BATCH) {
        int pos = atomicAdd(&cursors[r[i]], 1);
        perm[pos] = i;
    }
}

// ---------------- main WMMA kernel: one relation-tile per workgroup ----------------
// Y(256x16) = M(256x256) * D(256x16),  D[:,j] = ent[h_j] - ent[t_j]
// out[b_j] = -sqrt( sum_row (Y[row,j] + r_emb[row])^2 )

__global__ __launch_bounds__(256) void k_transr(
    const int* __restrict__ h, const int* __restrict__ t,
    const float* __restrict__ ent, const float* __restrict__ relE,
    const float* __restrict__ proj, const int* __restrict__ perm,
    const int* __restrict__ tiles, const int* __restrict__ ntiles,
    float* __restrict__ out) {

    __shared__ float dcol[16 * COL_STRIDE];   // D staged column-major, padded
    __shared__ float partial[16][32];         // per-(row-tile, lane) sum of squares

    const int g = blockIdx.x;
    if (g >= *ntiles) return;                 // uniform across wave

    const int rel   = tiles[g * 3 + 0];
    const int start = tiles[g * 3 + 1];
    const int len   = tiles[g * 3 + 2];
    const int tid   = threadIdx.x;

    // Build D: thread tid supplies element k=tid of every column (coalesced rows).
    for (int j = 0; j < 16; ++j) {
        float v = 0.0f;
        if (j < len) {
            int b = perm[start + j];
            v = ent[(size_t)h[b] * DIM + tid] - ent[(size_t)t[b] * DIM + tid];
        }
        dcol[j * COL_STRIDE + tid] = v;
    }
    __syncthreads();

    const int wave = tid >> 5, lane = tid & 31;
    const int col  = lane & 15, half = lane >> 4;  // A-row / B-col = lane%16, K-half = lane/16

    const float* Mb     = proj + (size_t)rel * (DIM * DIM);
    const float* relRow = relE + (size_t)rel * DIM;

    // 8 waves x 2 row-tiles = 16 tiles = 256 output rows
    for (int rt = 0; rt < 2; ++rt) {
        const int ro = (wave * 2 + rt) * 16;
        // A fragment source: row = ro + lane%16, K offset = half*2 (ISA 16x4 A layout)
        const float* arow = Mb + (size_t)(ro + col) * DIM + half * 2;
        const float* bptr = &dcol[col * COL_STRIDE + half * 2];

        v8f acc = {0.f, 0.f, 0.f, 0.f, 0.f, 0.f, 0.f, 0.f};
#pragma unroll 8
        for (int kk = 0; kk < DIM; kk += 4) {
            v2f a  = *(const v2f*)(arow + kk);   // M[ro+col][kk+half*2 .. +1]
            v2f bb = *(const v2f*)(bptr + kk);   // D[kk+half*2 .. +1][col]
            acc = __builtin_amdgcn_wmma_f32_16x16x4_f32(
                false, a, false, bb, (short)0, acc, false, false);
        }

        // diff = y + r_emb; per-lane partial over its 8 rows (fixed order)
        float s = 0.0f;
#pragma unroll
        for (int v = 0; v < 8; ++v) {
            int row = ro + v + half * 8;         // C/D layout: VGPR v -> M=v (+8 upper half)
            float diff = acc[v] + relRow[row];
            s += diff * diff;
        }
        partial[wave * 2 + rt][lane] = s;
    }
    __syncthreads();

    // Fixed-order reduction per column -> deterministic output per batch element.
    if (tid < len) {
        float s = 0.0f;
#pragma unroll
        for (int tt = 0; tt < 16; ++tt)
            s += partial[tt][tid] + partial[tt][tid + 16];
        out[perm[start + tid]] = -sqrtf(s);
    }
}

// ---------------- launch ----------------

extern "C" void kernel_launch(void* const* d_in, const int* in_sizes, int n_in,
                              void* d_out, int out_size, void* d_ws, size_t ws_size,
                              hipStream_t stream) {
    const int*   h    = (const int*)d_in[0];
    const int*   r    = (const int*)d_in[1];
    const int*   t    = (const int*)d_in[2];
    const float* ent  = (const float*)d_in[3];
    const float* relE = (const float*)d_in[4];
    const float* proj = (const float*)d_in[5];
    float*       out  = (float*)d_out;

    int* w       = (int*)d_ws;
    int* counts  = w;                       // 1000
    int* cursors = w + NUM_REL;             // 1000
    int* perm    = w + 2 * NUM_REL;         // 8192
    int* tiles   = w + 2 * NUM_REL + BATCH; // 3 * 1512
    int* ntiles  = tiles + 3 * MAX_TILES;   // 1

    k_zero   <<<(NUM_REL + 255) / 256, 256, 0, stream>>>(counts, ntiles);
    k_hist   <<<(BATCH   + 255) / 256, 256, 0, stream>>>(r, counts);
    k_build  <<<1, 1, 0, stream>>>(counts, cursors, tiles, ntiles);
    k_scatter<<<(BATCH   + 255) / 256, 256, 0, stream>>>(r, cursors, perm);
    k_transr <<<MAX_TILES, 256, 0, stream>>>(h, t, ent, relE, proj, perm,
                                             tiles, ntiles, out);
}